// VariationalLinearEncoder_29721173688331
// MI455X (gfx1250) — compile-verified
//
#include <hip/hip_runtime.h>
#include <hip/hip_bf16.h>

typedef __attribute__((ext_vector_type(2))) float v2f;
typedef __attribute__((ext_vector_type(8))) float v8f;

#define NNODES 100000
#define IN_C   256
#define OUT_C  64
#define NEDGES 3200000
#define MTILES (NNODES / 16)   // 6250, exact

// ---------------------------------------------------------------------------
// GEMM: h_mu = x @ W_mu, h_ls = x @ W_ls using V_WMMA_F32_16X16X4_F32.
// One wave32 -> one 16-row x 64-col strip of BOTH outputs (A loaded once).
// A (16x4 f32): lanes 0-15 hold K=0,1 ; lanes 16-31 hold K=2,3 (ISA layout).
// B (4x16 f32): VGPR0 -> K=0 (lanes 0-15) / K=2 (lanes 16-31); VGPR1 -> K=1/K=3.
// C/D (16x16 f32): vgpr v, lanes 0-15 -> row v, lanes 16-31 -> row v+8.
// ---------------------------------------------------------------------------
__global__ __launch_bounds__(256) void gcn_gemm_wmma(
    const float* __restrict__ x,
    const float* __restrict__ Wmu, const float* __restrict__ Wls,
    float* __restrict__ hmu, float* __restrict__ hls)
{
  const int wave = (blockIdx.x * blockDim.x + threadIdx.x) >> 5;
  if (wave >= MTILES) return;                    // wave-uniform: EXEC stays all-1
  const int lane  = threadIdx.x & 31;
  const int lrow  = lane & 15;
  const int khalf = (lane >> 4) << 1;            // 0 for lanes 0-15, 2 for 16-31
  const int m0    = wave << 4;

  const float* arow = x + (size_t)(m0 + lrow) * IN_C + khalf;

  v8f accmu[4] = {v8f{}, v8f{}, v8f{}, v8f{}};
  v8f accls[4] = {v8f{}, v8f{}, v8f{}, v8f{}};

  for (int k0 = 0; k0 < IN_C; k0 += 4) {
    const v2f a = *(const v2f*)(arow + k0);      // {K=khalf, K=khalf+1}, 8B aligned
    const float* bm = Wmu + (size_t)(k0 + khalf) * OUT_C + lrow;
    const float* bl = Wls + (size_t)(k0 + khalf) * OUT_C + lrow;
#pragma unroll
    for (int nt = 0; nt < 4; ++nt) {
      v2f b;
      b.x = bm[nt * 16];
      b.y = bm[nt * 16 + OUT_C];
      accmu[nt] = __builtin_amdgcn_wmma_f32_16x16x4_f32(
          false, a, false, b, (short)0, accmu[nt], false, false);
      b.x = bl[nt * 16];
      b.y = bl[nt * 16 + OUT_C];
      accls[nt] = __builtin_amdgcn_wmma_f32_16x16x4_f32(
          false, a, false, b, (short)0, accls[nt], false, false);
    }
  }

  const int rbase = m0 + ((lane >> 4) << 3);     // +8 for lanes 16-31
#pragma unroll
  for (int nt = 0; nt < 4; ++nt) {
    const int col = nt * 16 + lrow;
#pragma unroll
    for (int v = 0; v < 8; ++v) {
      hmu[(size_t)(rbase + v) * OUT_C + col] = accmu[nt][v];
      hls[(size_t)(rbase + v) * OUT_C + col] = accls[nt][v];
    }
  }
}

// deg[i] = 1 (self-loop contribution)
__global__ __launch_bounds__(256) void deg_init(float* __restrict__ deg)
{
  const int i = blockIdx.x * blockDim.x + threadIdx.x;
  if (i < NNODES) deg[i] = 1.0f;
}

// deg[dst[e]] += 1 for every real edge
__global__ __launch_bounds__(256) void deg_count(
    const int* __restrict__ dst, float* __restrict__ deg)
{
  const int e = blockIdx.x * blockDim.x + threadIdx.x;
  if (e < NEDGES) unsafeAtomicAdd(&deg[dst[e]], 1.0f);
}

// dinv = rsqrt(deg); out = dinv^2 * h (self-loop term) + bias. One wave per node.
__global__ __launch_bounds__(256) void node_init(
    const float* __restrict__ deg, float* __restrict__ dinv,
    const float* __restrict__ hmu, const float* __restrict__ hls,
    const float* __restrict__ bmu, const float* __restrict__ bls,
    float* __restrict__ omu, float* __restrict__ ols)
{
  const int gid  = blockIdx.x * blockDim.x + threadIdx.x;
  const int i    = gid >> 5;
  const int lane = gid & 31;
  if (i >= NNODES) return;
  const float di = __frsqrt_rn(deg[i]);          // deg >= 1 always (self-loop)
  if (lane == 0) dinv[i] = di;
  const float w = di * di;
  const int c = lane << 1;                       // 2 channels per lane
  const v2f hm = *(const v2f*)(hmu + (size_t)i * OUT_C + c);
  const v2f hl = *(const v2f*)(hls + (size_t)i * OUT_C + c);
  const v2f bm = *(const v2f*)(bmu + c);
  const v2f bl = *(const v2f*)(bls + c);
  v2f om; om.x = w * hm.x + bm.x; om.y = w * hm.y + bm.y;
  v2f ol; ol.x = w * hl.x + bl.x; ol.y = w * hl.y + bl.y;
  *(v2f*)(omu + (size_t)i * OUT_C + c) = om;
  *(v2f*)(ols + (size_t)i * OUT_C + c) = ol;
}

// One wave32 per edge: gather 64+64 channels from h_mu/h_ls (L2-resident),
// scale by dinv[s]*dinv[d], scatter-add via hardware f32 atomics.
__global__ __launch_bounds__(256) void edge_scatter(
    const int* __restrict__ src, const int* __restrict__ dst,
    const float* __restrict__ dinv,
    const float* __restrict__ hmu, const float* __restrict__ hls,
    float* __restrict__ omu, float* __restrict__ ols)
{
  const int gid  = blockIdx.x * blockDim.x + threadIdx.x;  // < 102.4M, fits int
  const int e    = gid >> 5;
  const int lane = gid & 31;
  if (e >= NEDGES) return;
  const int s = src[e];
  const int d = dst[e];
  const float w = dinv[s] * dinv[d];
  const int c = lane << 1;
  const v2f hm = *(const v2f*)(hmu + (size_t)s * OUT_C + c);
  const v2f hl = *(const v2f*)(hls + (size_t)s * OUT_C + c);
  float* pm = omu + (size_t)d * OUT_C + c;
  float* pl = ols + (size_t)d * OUT_C + c;
  unsafeAtomicAdd(pm,     w * hm.x);
  unsafeAtomicAdd(pm + 1, w * hm.y);
  unsafeAtomicAdd(pl,     w * hl.x);
  unsafeAtomicAdd(pl + 1, w * hl.y);
}

extern "C" void kernel_launch(void* const* d_in, const int* in_sizes, int n_in,
                              void* d_out, int out_size, void* d_ws, size_t ws_size,
                              hipStream_t stream)
{
  const float* x   = (const float*)d_in[0];
  const int*   ei  = (const int*)d_in[1];       // [2, E] indices
  const float* Wmu = (const float*)d_in[2];
  const float* bmu = (const float*)d_in[3];
  const float* Wls = (const float*)d_in[4];
  const float* bls = (const float*)d_in[5];
  (void)in_sizes; (void)n_in; (void)out_size; (void)ws_size;

  const int* src = ei;
  const int* dst = ei + NEDGES;

  float* omu = (float*)d_out;
  float* ols = omu + (size_t)NNODES * OUT_C;

  // workspace: h_mu (25.6MB) | h_ls (25.6MB) | deg (0.4MB) | dinv (0.4MB)
  float* hmu  = (float*)d_ws;
  float* hls  = hmu + (size_t)NNODES * OUT_C;
  float* deg  = hls + (size_t)NNODES * OUT_C;
  float* dinv = deg + NNODES;

  gcn_gemm_wmma<<<(MTILES * 32 + 255) / 256, 256, 0, stream>>>(x, Wmu, Wls, hmu, hls);
  deg_init    <<<(NNODES + 255) / 256,       256, 0, stream>>>(deg);
  deg_count   <<<(NEDGES + 255) / 256,       256, 0, stream>>>(dst, deg);
  node_init   <<<(NNODES * 32 + 255) / 256,  256, 0, stream>>>(deg, dinv, hmu, hls,
                                                               bmu, bls, omu, ols);
  const long long escatter_threads = (long long)NEDGES * 32;
  edge_scatter<<<(int)((escatter_threads + 255) / 256), 256, 0, stream>>>(
      src, dst, dinv, hmu, hls, omu, ols);
}